// SAGEGRU_18700287606789
// MI455X (gfx1250) — compile-verified
//
#include <hip/hip_runtime.h>
#include <hip/hip_bf16.h>
#include <stdint.h>

#define B_    16
#define T_    64
#define NN    5000
#define NPAD  5008   // multiple of 16
#define EE    80000
#define HG_   64
#define HT_   128

typedef float v2f __attribute__((ext_vector_type(2)));
typedef float v8f __attribute__((ext_vector_type(8)));

// ---------------- setup kernels ----------------

__global__ void zero_f32(float* p, int n) {
    int i = blockIdx.x * blockDim.x + threadIdx.x;
    if (i < n) p[i] = 0.f;
}

__global__ void zero_i32(int* p, int n) {
    int i = blockIdx.x * blockDim.x + threadIdx.x;
    if (i < n) p[i] = 0;
}

__global__ void count_deg(const int* ei, int* deg) {
    int e = blockIdx.x * blockDim.x + threadIdx.x;
    if (e < EE) atomicAdd(&deg[ei[EE + e]], 1);
}

// single-block exclusive scan over deg -> row_ptr, cursor, inv_deg
__global__ void scan_deg(const int* deg, int* row_ptr, int* cursor, float* inv_deg) {
    __shared__ int sm[1024];
    __shared__ int carry;
    if (threadIdx.x == 0) { carry = 0; row_ptr[0] = 0; }
    __syncthreads();
    for (int base = 0; base < NN; base += 1024) {
        int i = base + (int)threadIdx.x;
        int v = (i < NN) ? deg[i] : 0;
        sm[threadIdx.x] = v;
        __syncthreads();
        for (int off = 1; off < 1024; off <<= 1) {
            int t = (threadIdx.x >= (unsigned)off) ? sm[threadIdx.x - off] : 0;
            __syncthreads();
            sm[threadIdx.x] += t;
            __syncthreads();
        }
        int incl = sm[threadIdx.x] + carry;
        if (i < NN) {
            row_ptr[i + 1] = incl;
            cursor[i]      = incl - v;  // exclusive prefix
            inv_deg[i]     = 1.0f / fmaxf((float)v, 1.0f);
        }
        __syncthreads();
        if (threadIdx.x == 1023) carry = incl;
        __syncthreads();
    }
    for (int i = NN + (int)threadIdx.x; i < NPAD; i += 1024) inv_deg[i] = 0.f;
}

__global__ void fill_csr(const int* ei, int* cursor, int* csr) {
    int e = blockIdx.x * blockDim.x + threadIdx.x;
    if (e < EE) {
        int d = ei[EE + e];
        int p = atomicAdd(&cursor[d], 1);
        csr[p] = ei[e];  // src node
    }
}

// ---------------- per-timestep SAGE kernels ----------------

// agg1[b][n] = inv_deg[n] * sum_{src in row n} x[b,t,src]   (F == 1)
__global__ void sage1_agg(const float* __restrict__ x, const int* __restrict__ rp,
                          const int* __restrict__ csr, const float* __restrict__ inv_deg,
                          float* __restrict__ agg1, int t) {
    int idx = blockIdx.x * blockDim.x + threadIdx.x;
    if (idx >= B_ * NPAD) return;
    int b = idx / NPAD, n = idx % NPAD;
    float s = 0.f;
    if (n < NN) {
        const float* xb = x + ((size_t)b * T_ + t) * NN;
        int e0 = rp[n], e1 = rp[n + 1];
        for (int j = e0; j < e1; ++j) s += xb[csr[j]];
        s *= inv_deg[n];
    }
    agg1[idx] = s;
}

// h1[row][f] = relu(agg1[row]*Wl[f] + x[b,t,n]*Wr[f] + b[f]); pad rows -> 0
__global__ void sage1_ffn(const float* __restrict__ x, const float* __restrict__ agg1,
                          const float* __restrict__ Wl, const float* __restrict__ Wr,
                          const float* __restrict__ bias, float* __restrict__ h1, int t) {
    int row = blockIdx.x * 4 + threadIdx.y;
    int f   = threadIdx.x;
    if (row >= B_ * NPAD) return;
    int b = row / NPAD, n = row % NPAD;
    float v = 0.f;
    if (n < NN) {
        float xa = x[((size_t)b * T_ + t) * NN + n];
        v = fmaxf(agg1[row] * Wl[f] + xa * Wr[f] + bias[f], 0.f);
    }
    h1[(size_t)row * HG_ + f] = v;
}

// Fused sage2: CSR aggregate -> LDS tile, WMMA GEMM (agg@Wl + h1@Wr + b),
// ReLU, mean over nodes -> outs[t]. Weights staged to LDS with CDNA5 async
// global->LDS loads (ASYNCcnt). One block = one 16-node tile x one batch.
__global__ void __launch_bounds__(128) sage2_fused(const float* __restrict__ h1,
                                                   const int* __restrict__ rp,
                                                   const int* __restrict__ csr,
                                                   const float* __restrict__ inv_deg,
                                                   const float* __restrict__ Wl,
                                                   const float* __restrict__ Wr,
                                                   const float* __restrict__ bias,
                                                   float* __restrict__ outs, int t) {
    __shared__ __align__(16) float lds_w[2 * HG_ * HG_];  // [Wl ; Wr], [k][col], 32 KB
    __shared__ __align__(16) float lds_a[16 * HG_];       // agg tile, 4 KB

    int tid  = threadIdx.x;
    int tile = blockIdx.x;            // 0..312
    int b    = blockIdx.y;            // batch
    int nbase = tile * 16;
    size_t hb = (size_t)b * NPAD * HG_;

    // ---- stage 0: async-copy Wl|Wr into LDS (2048 x b128) ----
    {
        unsigned lbase = (unsigned)(uintptr_t)(&lds_w[0]);
#pragma unroll
        for (int i = 0; i < 16; ++i) {
            int e4 = tid + i * 128;                   // float4 index 0..2047
            unsigned loff = lbase + (unsigned)e4 * 16u;
            const float* g = (e4 < HG_ * HG_ / 4) ? (Wl + (size_t)e4 * 4)
                                                  : (Wr + (size_t)(e4 - HG_ * HG_ / 4) * 4);
            unsigned long long ga = (unsigned long long)(uintptr_t)g;
#if defined(__gfx1250__)
            asm volatile("global_load_async_to_lds_b128 %0, %1, off"
                         :: "v"(loff), "v"(ga) : "memory");
#else
            // host pass / non-gfx1250 fallback: plain copy
            *(float4*)((char*)lds_w + (e4 * 16)) = *(const float4*)g;
#endif
        }
#if defined(__gfx1250__)
        asm volatile("s_wait_asynccnt 0" ::: "memory");
#endif
    }

    // ---- stage 1: CSR-aggregate 16x64 tile into LDS (uniform per wave) ----
    {
        int f  = tid & 63;
        int y2 = tid >> 6;  // 0..1
#pragma unroll
        for (int r = 0; r < 8; ++r) {
            int row = y2 * 8 + r;
            int n   = nbase + row;
            float s = 0.f;
            if (n < NN) {
                int e0 = rp[n], e1 = rp[n + 1];
                for (int j = e0; j < e1; ++j) s += h1[hb + (size_t)csr[j] * HG_ + f];
                s *= inv_deg[n];
            }
            lds_a[row * HG_ + f] = s;
        }
    }
    __syncthreads();

    // ---- stage 2: WMMA fp32 16x16x4, K=64, two matrices ----
    int wave = tid >> 5;              // col tile 0..3
    int lane = tid & 31;
    int m    = lane & 15;
    int half = lane >> 4;
    int kk   = half * 2;
    int col  = wave * 16 + m;
    const float* A2 = h1 + hb + (size_t)nbase * HG_;

    v8f c;
    float bv = bias[col];
#pragma unroll
    for (int v = 0; v < 8; ++v) c[v] = bv;

#if defined(__gfx1250__) && __has_builtin(__builtin_amdgcn_wmma_f32_16x16x4_f32)
#pragma unroll
    for (int kb = 0; kb < HG_; kb += 4) {
        v2f a1 = *(const v2f*)&lds_a[m * HG_ + kb + kk];
        v2f b1v; b1v.x = lds_w[(kb + kk) * HG_ + col];
                 b1v.y = lds_w[(kb + kk + 1) * HG_ + col];
        c = __builtin_amdgcn_wmma_f32_16x16x4_f32(false, a1, false, b1v, (short)0, c, false, false);
        v2f a2 = *(const v2f*)(A2 + (size_t)m * HG_ + kb + kk);
        v2f b2v; b2v.x = lds_w[HG_ * HG_ + (kb + kk) * HG_ + col];
                 b2v.y = lds_w[HG_ * HG_ + (kb + kk + 1) * HG_ + col];
        c = __builtin_amdgcn_wmma_f32_16x16x4_f32(false, a2, false, b2v, (short)0, c, false, false);
    }
#else
    // scalar fallback with identical C fragment layout
#pragma unroll
    for (int v = 0; v < 8; ++v) {
        int mm = half * 8 + v;
        float acc = c[v];
        for (int k = 0; k < HG_; ++k)
            acc += lds_a[mm * HG_ + k] * lds_w[k * HG_ + col]
                 + A2[(size_t)mm * HG_ + k] * lds_w[HG_ * HG_ + k * HG_ + col];
        c[v] = acc;
    }
#endif

    // ---- stage 3: relu + masked mean -> one atomic per lane ----
    float partial = 0.f;
#pragma unroll
    for (int v = 0; v < 8; ++v) {
        int n = nbase + half * 8 + v;   // C layout: VGPR v holds M = v (+8 upper half)
        float val = fmaxf(c[v], 0.f);
        if (n < NN) partial += val;
    }
    atomicAdd(&outs[((size_t)t * B_ + b) * HG_ + col], partial * (1.0f / (float)NN));
}

// ---------------- GRU + head (single workgroup, LDS-resident state) ----------------

__global__ void gru_head(const float* __restrict__ outs,
                         const float* __restrict__ Wih, const float* __restrict__ Whh,
                         const float* __restrict__ bih, const float* __restrict__ bhh,
                         const float* __restrict__ h1W, const float* __restrict__ h1b,
                         const float* __restrict__ h2W, const float* __restrict__ h2b,
                         float* __restrict__ out) {
    __shared__ float h [B_ * HT_];        // 8 KB
    __shared__ float gi[B_ * 3 * HT_];    // 24 KB
    __shared__ float gh[B_ * 3 * HT_];    // 24 KB
    __shared__ float xt[B_ * HG_];        // 4 KB
    int tid = threadIdx.x;
    for (int i = tid; i < B_ * HT_; i += blockDim.x) h[i] = 0.f;
    __syncthreads();

    for (int t = 0; t < T_; ++t) {
        for (int i = tid; i < B_ * HG_; i += blockDim.x)
            xt[i] = outs[(size_t)t * B_ * HG_ + i];
        __syncthreads();
        for (int i = tid; i < B_ * 3 * HT_; i += blockDim.x) {
            int b = i / (3 * HT_), g = i % (3 * HT_);
            const float* w  = Wih + (size_t)g * HG_;
            const float* xb = xt + b * HG_;
            float s = 0.f;
            for (int k = 0; k < HG_; ++k) s += xb[k] * w[k];
            gi[i] = s + bih[g];
            const float* w2 = Whh + (size_t)g * HT_;
            const float* hbp = h + b * HT_;
            float s2 = 0.f;
            for (int k = 0; k < HT_; ++k) s2 += hbp[k] * w2[k];
            gh[i] = s2 + bhh[g];
        }
        __syncthreads();
        for (int i = tid; i < B_ * HT_; i += blockDim.x) {
            int b = i / HT_, u = i % HT_;
            float ir = gi[b * 3 * HT_ + u],         hr  = gh[b * 3 * HT_ + u];
            float iz = gi[b * 3 * HT_ + HT_ + u],   hz  = gh[b * 3 * HT_ + HT_ + u];
            float in_= gi[b * 3 * HT_ + 2*HT_ + u], hnn = gh[b * 3 * HT_ + 2*HT_ + u];
            float r = 1.f / (1.f + expf(-(ir + hr)));
            float z = 1.f / (1.f + expf(-(iz + hz)));
            float nn = tanhf(in_ + r * hnn);
            h[i] = (1.f - z) * nn + z * h[i];   // in-place: each thread owns i
        }
        __syncthreads();
    }

    if (tid < B_) {
        int b = tid;
        float acc = h2b[0];
        for (int j = 0; j < 64; ++j) {
            float s = h1b[j];
            for (int k = 0; k < HT_; ++k) s += h[b * HT_ + k] * h1W[(size_t)k * 64 + j];
            acc += fmaxf(s, 0.f) * h2W[j];
        }
        out[b] = acc;
    }
}

// ---------------- launcher ----------------

extern "C" void kernel_launch(void* const* d_in, const int* in_sizes, int n_in,
                              void* d_out, int out_size, void* d_ws, size_t ws_size,
                              hipStream_t stream) {
    const float* x     = (const float*)d_in[0];
    const int*   ei    = (const int*)  d_in[1];
    const float* s1_Wl = (const float*)d_in[2];
    const float* s1_Wr = (const float*)d_in[3];
    const float* s1_b  = (const float*)d_in[4];
    const float* s2_Wl = (const float*)d_in[5];
    const float* s2_Wr = (const float*)d_in[6];
    const float* s2_b  = (const float*)d_in[7];
    const float* gWih  = (const float*)d_in[8];
    const float* gWhh  = (const float*)d_in[9];
    const float* gbih  = (const float*)d_in[10];
    const float* gbhh  = (const float*)d_in[11];
    const float* h1W   = (const float*)d_in[12];
    const float* h1b   = (const float*)d_in[13];
    const float* h2W   = (const float*)d_in[14];
    const float* h2b   = (const float*)d_in[15];
    float* out = (float*)d_out;

    char* p = (char*)d_ws;
    float* inv_deg = (float*)p; p += (size_t)NPAD * 4;
    float* agg1    = (float*)p; p += (size_t)B_ * NPAD * 4;
    float* h1      = (float*)p; p += (size_t)B_ * NPAD * HG_ * 4;
    float* outs    = (float*)p; p += (size_t)T_ * B_ * HG_ * 4;
    int*   deg     = (int*)p;   p += (size_t)NN * 4;
    int*   rp      = (int*)p;   p += (size_t)(NN + 1) * 4;
    int*   cursor  = (int*)p;   p += (size_t)NN * 4;
    int*   csr     = (int*)p;   p += (size_t)EE * 4;
    if ((size_t)(p - (char*)d_ws) > ws_size) return;

    // CSR build (graph is time-invariant)
    zero_i32 <<<(NN + 255) / 256, 256, 0, stream>>>(deg, NN);
    count_deg<<<(EE + 255) / 256, 256, 0, stream>>>(ei, deg);
    scan_deg <<<1, 1024, 0, stream>>>(deg, rp, cursor, inv_deg);
    fill_csr <<<(EE + 255) / 256, 256, 0, stream>>>(ei, cursor, csr);
    zero_f32 <<<(T_ * B_ * HG_ + 255) / 256, 256, 0, stream>>>(outs, T_ * B_ * HG_);

    for (int t = 0; t < T_; ++t) {
        sage1_agg<<<(B_ * NPAD + 255) / 256, 256, 0, stream>>>(x, rp, csr, inv_deg, agg1, t);
        sage1_ffn<<<dim3(B_ * NPAD / 4), dim3(64, 4), 0, stream>>>(x, agg1, s1_Wl, s1_Wr, s1_b, h1, t);
        sage2_fused<<<dim3(NPAD / 16, B_), 128, 0, stream>>>(h1, rp, csr, inv_deg,
                                                             s2_Wl, s2_Wr, s2_b, outs, t);
    }

    gru_head<<<1, 512, 0, stream>>>(outs, gWih, gWhh, gbih, gbhh, h1W, h1b, h2W, h2b, out);
}